// DifferentiableOrthogonalMatchingPursuit_40432822125179
// MI455X (gfx1250) — compile-verified
//
#include <hip/hip_runtime.h>
#include <stdint.h>

// ---------------------------------------------------------------------------
// Differentiable OMP forward for MI455X (gfx1250, wave32).
//  - proj  = R @ D   : M=64, N=1024, K=1024 f32 GEMM via V_WMMA_F32_16X16X4_F32
//    * A-tile (16x1024 residuals) staged into LDS with ASYNC global->LDS copies
//      (K-major layout => conflict-free ds_read_b32 for WMMA A-fragments)
//    * B streamed from global (4MB dictionary, L2-resident) + global_prefetch_b8
//  - argmax fused into GEMM epilogue via packed u64 global_atomic_max
//  - Gram  = Sel Sel^T : k x k x 1024 f32 GEMM via the same WMMA op
//  - tiny Cholesky solve per batch, residual update, sparse recon
// ---------------------------------------------------------------------------

typedef __attribute__((ext_vector_type(2))) float v2f;
typedef __attribute__((ext_vector_type(8))) float v8f;

#define B_N   64
#define L_N   1024
#define A_N   1024
#define K_SEL 32
// softplus(-5) = log1p(exp(-5))
#define REG_F 0.0067153485f

// ------------------- CDNA5 async global->LDS helpers ------------------------
#if __has_builtin(__builtin_amdgcn_global_load_async_to_lds_b32)
#define OMP_ASYNC_LDS 1
#endif

typedef __attribute__((address_space(1))) int g1_int;  // global (AS1)
typedef __attribute__((address_space(3))) int l3_int;  // LDS (AS3)

__device__ __forceinline__ void omp_async_wait0() {
#if __has_builtin(__builtin_amdgcn_s_wait_asynccnt)
    __builtin_amdgcn_s_wait_asynccnt(0);
#else
    asm volatile("s_wait_asynccnt 0x0" ::: "memory");
#endif
}

__device__ __forceinline__ void omp_copy_g2l_b32(const float* g, float* l) {
#if defined(OMP_ASYNC_LDS)
    __builtin_amdgcn_global_load_async_to_lds_b32((g1_int*)g, (l3_int*)l, 0, 0);
#else
    *l = *g;
#endif
}

// --------------------------- helpers ---------------------------------------
__device__ inline unsigned long long pack_am(float p, unsigned a) {
    unsigned fb = __float_as_uint(fabsf(p));
    // monotone in |p|; low bits favor the SMALLEST atom index on ties
    return ((unsigned long long)fb << 32) | (unsigned long long)(0xFFFFFFFFu - a);
}

__device__ inline unsigned long long shflxor_u64_w16(unsigned long long v, int mask) {
    int lo = __shfl_xor((int)(unsigned)(v & 0xFFFFFFFFull), mask, 16);
    int hi = __shfl_xor((int)(unsigned)(v >> 32), mask, 16);
    return ((unsigned long long)(unsigned)hi << 32) | (unsigned long long)(unsigned)lo;
}

// --------------------------- kernels ---------------------------------------

// R = y  (residual init)
__global__ void omp_init_kernel(const float* __restrict__ y, float* __restrict__ R) {
    int t = blockIdx.x * blockDim.x + threadIdx.x;
    if (t < B_N * L_N) R[t] = y[t];
}

__global__ void omp_zero_argmax(unsigned long long* __restrict__ am) {
    if (threadIdx.x < B_N) am[threadIdx.x] = 0ull;
}

// P = R(64x1024) @ D(1024x1024) with f32 WMMA 16x16x4; fused |.| argmax.
// grid (4,16), block 128 (4 waves). wave w -> tile (m0=bx*16, n0=(by*4+w)*16)
__global__ void omp_proj_wmma(const float* __restrict__ R, const float* __restrict__ X0,
                              unsigned long long* __restrict__ am) {
    // A-tile staged K-major: sA[k*16 + r] = R[m0+r][k]   (exactly 64 KB)
    __shared__ float sA[L_N * 16];

    const int tid  = threadIdx.x;
    const int lane = tid & 31;
    const int wave = tid >> 5;
    const int m0   = blockIdx.x * 16;
    const int n0   = (blockIdx.y * 4 + wave) * 16;
    const int half = lane >> 4;   // 0: K={0,1}, 1: K={2,3}
    const int lid  = lane & 15;

    // ---- cooperative async stage of the contiguous 64KB A-tile ----
    // rows m0..m0+15 of R are contiguous: flat element f goes to LDS float index (f%1024)*16 + f/1024
    {
        const float* gsrc = R + (size_t)m0 * L_N;
        #pragma unroll
        for (int j = 0; j < (16 * L_N) / 128; ++j) {
            const int f = tid + 128 * j;
            omp_copy_g2l_b32(gsrc + f, &sA[(f & (L_N - 1)) * 16 + (f >> 10)]);
        }
    }
    omp_async_wait0();
    __syncthreads();

    // A(16x4): lane lid holds row m=lid, ks = half*2 .. half*2+1 (from LDS)
    const float* sarow = &sA[(half * 2) * 16 + lid];
    // B(4x16): lane lid holds col n=lid, rows K=half*2, half*2+1  (stride A_N)
    const float* bcol = X0 + (size_t)(half * 2) * A_N + (n0 + lid);

    v8f c = {};
    for (int k = 0; k < L_N; k += 4) {
        v2f a, b;
        a.x = sarow[(size_t)k * 16];
        a.y = sarow[(size_t)k * 16 + 16];
        b.x = bcol[(size_t)k * A_N];
        b.y = bcol[(size_t)k * A_N + A_N];
        // prefetch dictionary rows ~16 K-steps ahead (global_prefetch_b8)
        __builtin_prefetch(&bcol[(size_t)(k + 16) * A_N], 0, 1);
        c = __builtin_amdgcn_wmma_f32_16x16x4_f32(false, a, false, b,
                                                  (short)0, c, false, false);
    }

    // epilogue: per-row |max| reduction across the 16 lanes holding one row,
    // then one packed u64 atomic max per (batch-row, wave).
    #pragma unroll
    for (int r = 0; r < 8; ++r) {
        unsigned n = (unsigned)(n0 + lid);
        unsigned long long p = pack_am(c[r], n);
        #pragma unroll
        for (int off = 1; off < 16; off <<= 1) {
            unsigned long long q = shflxor_u64_w16(p, off);
            if (q > p) p = q;
        }
        if (lid == 0) {
            int m = m0 + r + half * 8;
            atomicMax(&am[m], p);
        }
    }
}

// proj for the implicit ones-atom (a = 1024): row sum of R
__global__ void omp_ones_proj(const float* __restrict__ R,
                              unsigned long long* __restrict__ am) {
    __shared__ float red[256];
    const int b = blockIdx.x;
    float s = 0.0f;
    for (int l = threadIdx.x; l < L_N; l += 256) s += R[(size_t)b * L_N + l];
    red[threadIdx.x] = s;
    __syncthreads();
    for (int st = 128; st > 0; st >>= 1) {
        if ((int)threadIdx.x < st) red[threadIdx.x] += red[threadIdx.x + st];
        __syncthreads();
    }
    if (threadIdx.x == 0) atomicMax(&am[b], pack_am(red[0], A_N));
}

// decode argmax, record index history, gather selected dictionary column
__global__ void omp_select(const float* __restrict__ X0,
                           const unsigned long long* __restrict__ am,
                           float* __restrict__ S, int* __restrict__ idx_hist, int iter) {
    const int b = blockIdx.y;
    const int l = blockIdx.x * blockDim.x + threadIdx.x;
    const unsigned idx = 0xFFFFFFFFu - (unsigned)(am[b] & 0xFFFFFFFFull);
    if (blockIdx.x == 0 && threadIdx.x == 0) idx_hist[iter * B_N + b] = (int)idx;
    const float v = (idx < A_N) ? X0[(size_t)l * A_N + idx] : 1.0f;
    S[((size_t)b * K_SEL + iter) * L_N + l] = v;
}

// Gram: DTD = Sel Sel^T (k x k, K=1024) via WMMA f32 16x16x4; DTy on 5th wave.
// grid 64 (one block per batch), block 160 (5 waves).
__global__ void omp_gram_wmma(const float* __restrict__ S, const float* __restrict__ y,
                              float* __restrict__ DTD, float* __restrict__ DTy, int kcur) {
    const int b    = blockIdx.x;
    const float* Sb = S + (size_t)b * K_SEL * L_N;
    const int wave = threadIdx.x >> 5;
    const int lane = threadIdx.x & 31;

    if (wave < 4) {
        const int mt = (wave >> 1) * 16, nt = (wave & 1) * 16;
        const int half = lane >> 4, lid = lane & 15;
        const int mrow = mt + lid;
        const int nrow = nt + lid;
        const bool mok = mrow < kcur, nok = nrow < kcur;
        const float* ap = Sb + (size_t)mrow * L_N + half * 2;
        const float* bp = Sb + (size_t)nrow * L_N + half * 2;

        v8f c = {};
        for (int k = 0; k < L_N; k += 4) {
            v2f a, bb;
            a.x  = mok ? ap[k]     : 0.0f;
            a.y  = mok ? ap[k + 1] : 0.0f;
            bb.x = nok ? bp[k]     : 0.0f;
            bb.y = nok ? bp[k + 1] : 0.0f;
            c = __builtin_amdgcn_wmma_f32_16x16x4_f32(false, a, false, bb,
                                                      (short)0, c, false, false);
        }
        #pragma unroll
        for (int r = 0; r < 8; ++r) {
            const int m = mt + r + half * 8;
            const int n = nt + lid;
            if (m < kcur && n < kcur) {
                float v = c[r] + (m == n ? REG_F : 0.0f);
                DTD[((size_t)b * K_SEL + m) * K_SEL + n] = v;
            }
        }
    } else {
        // wave 4: DTy[u] = <Sel_u, y_b>
        const int u = lane;
        if (u < kcur) {
            const float* sp = Sb + (size_t)u * L_N;
            const float* yp = y  + (size_t)b * L_N;
            float acc = 0.0f;
            for (int l = 0; l < L_N; ++l) acc = fmaf(sp[l], yp[l], acc);
            DTy[b * K_SEL + u] = acc;
        }
    }
}

// per-batch Cholesky solve of (DTD) w = DTy, k <= 32
__global__ void omp_solve(const float* __restrict__ DTD, const float* __restrict__ DTy,
                          float* __restrict__ W, int kcur) {
    __shared__ float aM[32 * 33];
    __shared__ float rhs[32], zv[32], wv[32];
    const int b = blockIdx.x;
    for (int t = threadIdx.x; t < kcur * kcur; t += blockDim.x) {
        int r = t / kcur, cc = t % kcur;
        aM[r * 33 + cc] = DTD[((size_t)b * K_SEL + r) * K_SEL + cc];
    }
    if ((int)threadIdx.x < kcur) rhs[threadIdx.x] = DTy[b * K_SEL + threadIdx.x];
    __syncthreads();
    if (threadIdx.x == 0) {
        for (int cc = 0; cc < kcur; ++cc) {
            float d = aM[cc * 33 + cc];
            for (int t = 0; t < cc; ++t) d -= aM[cc * 33 + t] * aM[cc * 33 + t];
            d = sqrtf(fmaxf(d, 1e-30f));
            aM[cc * 33 + cc] = d;
            const float inv = 1.0f / d;
            for (int r = cc + 1; r < kcur; ++r) {
                float s = aM[r * 33 + cc];
                for (int t = 0; t < cc; ++t) s -= aM[r * 33 + t] * aM[cc * 33 + t];
                aM[r * 33 + cc] = s * inv;
            }
        }
        for (int r = 0; r < kcur; ++r) {
            float s = rhs[r];
            for (int t = 0; t < r; ++t) s -= aM[r * 33 + t] * zv[t];
            zv[r] = s / aM[r * 33 + r];
        }
        for (int r = kcur - 1; r >= 0; --r) {
            float s = zv[r];
            for (int t = r + 1; t < kcur; ++t) s -= aM[t * 33 + r] * wv[t];
            wv[r] = s / aM[r * 33 + r];
        }
    }
    __syncthreads();
    if ((int)threadIdx.x < kcur) W[b * K_SEL + threadIdx.x] = wv[threadIdx.x];
}

// R = y - Sel^T w
__global__ void omp_resid(const float* __restrict__ S, const float* __restrict__ y,
                          const float* __restrict__ W, float* __restrict__ R, int kcur) {
    __shared__ float wv[K_SEL];
    const int b = blockIdx.y;
    const int l = blockIdx.x * blockDim.x + threadIdx.x;
    if ((int)threadIdx.x < kcur) wv[threadIdx.x] = W[b * K_SEL + threadIdx.x];
    __syncthreads();
    float acc = y[(size_t)b * L_N + l];
    for (int u = 0; u < kcur; ++u)
        acc = fmaf(-wv[u], S[((size_t)b * K_SEL + u) * L_N + l], acc);
    R[(size_t)b * L_N + l] = acc;
}

// recon[b,l] = sum_i m_i * w_i * Dfull[b, l, idx_i]  (last-write-wins scatter)
__global__ void omp_recon(const float* __restrict__ X, const int* __restrict__ idx_hist,
                          const float* __restrict__ W, float* __restrict__ out) {
    __shared__ float wv[K_SEL];
    __shared__ int   idxs[K_SEL];
    const int b = blockIdx.y;
    const int l = blockIdx.x * blockDim.x + threadIdx.x;
    if (threadIdx.x < K_SEL) {
        const int my = idx_hist[threadIdx.x * B_N + b];
        idxs[threadIdx.x] = my;
        float w = W[b * K_SEL + threadIdx.x];
        for (int j = threadIdx.x + 1; j < K_SEL; ++j)
            if (idx_hist[j * B_N + b] == my) { w = 0.0f; break; }
        wv[threadIdx.x] = w;
    }
    __syncthreads();
    const float* Xb = X + (size_t)b * L_N * A_N + (size_t)l * A_N;
    float acc = 0.0f;
    #pragma unroll 4
    for (int i = 0; i < K_SEL; ++i) {
        const int a = idxs[i];
        const float dv = (a < A_N) ? Xb[a] : 1.0f;
        acc = fmaf(wv[i], dv, acc);
    }
    out[(size_t)b * L_N + l] = acc;
}

// --------------------------- launcher ---------------------------------------
extern "C" void kernel_launch(void* const* d_in, const int* in_sizes, int n_in,
                              void* d_out, int out_size, void* d_ws, size_t ws_size,
                              hipStream_t stream) {
    const float* X = (const float*)d_in[0];  // (64,1024,1024); X[0] is the dictionary
    const float* y = (const float*)d_in[1];  // (64,1024,1)
    float* out = (float*)d_out;              // (64,1024,1)

    char* ws = (char*)d_ws;
    size_t off = 0;
    auto carve = [&](size_t bytes) -> void* {
        void* p = ws + off;
        off = (off + bytes + 255) & ~(size_t)255;
        return p;
    };
    float* R   = (float*)carve((size_t)B_N * L_N * 4);                // residuals
    float* S   = (float*)carve((size_t)B_N * K_SEL * L_N * 4);        // selected cols (row-major per atom)
    float* DTD = (float*)carve((size_t)B_N * K_SEL * K_SEL * 4);
    float* DTy = (float*)carve((size_t)B_N * K_SEL * 4);
    float* W   = (float*)carve((size_t)B_N * K_SEL * 4);
    unsigned long long* am = (unsigned long long*)carve((size_t)B_N * 8);
    int* ih    = (int*)carve((size_t)K_SEL * B_N * 4);
    (void)in_sizes; (void)n_in; (void)out_size; (void)ws_size;

    omp_init_kernel<<<(B_N * L_N) / 256, 256, 0, stream>>>(y, R);
    for (int i = 0; i < K_SEL; ++i) {
        omp_zero_argmax<<<1, 64, 0, stream>>>(am);
        omp_proj_wmma<<<dim3(4, 16), 128, 0, stream>>>(R, X, am);
        omp_ones_proj<<<B_N, 256, 0, stream>>>(R, am);
        omp_select<<<dim3(4, B_N), 256, 0, stream>>>(X, am, S, ih, i);
        omp_gram_wmma<<<B_N, 160, 0, stream>>>(S, y, DTD, DTy, i + 1);
        omp_solve<<<B_N, 64, 0, stream>>>(DTD, DTy, W, i + 1);
        omp_resid<<<dim3(4, B_N), 256, 0, stream>>>(S, y, W, R, i + 1);
    }
    omp_recon<<<dim3(4, B_N), 256, 0, stream>>>(X, ih, W, out);
}